// GruAttentionSingleCritic_75350906241224
// MI455X (gfx1250) — compile-verified
//
#include <hip/hip_runtime.h>

typedef float v2f __attribute__((ext_vector_type(2)));
typedef float v4f __attribute__((ext_vector_type(4)));
typedef float v8f __attribute__((ext_vector_type(8)));
typedef int   v4i __attribute__((ext_vector_type(4)));

#define A_DIM   16
#define B_DIM   65536
#define S_DIM   64
#define HID     32

#define LRELU(x) ((x) > 0.0f ? (x) : 0.01f*(x))

#if __has_builtin(__builtin_amdgcn_global_load_async_to_lds_b128)
#define USE_ASYNC_LDS 1
typedef __attribute__((address_space(1))) v4i* GV4P;   // global-side (prints as __device__)
typedef __attribute__((address_space(3))) v4i* LV4P;   // LDS-side
#endif

__device__ __forceinline__ void wait_async0() {
#if defined(USE_ASYNC_LDS)
#if __has_builtin(__builtin_amdgcn_s_wait_asynccnt)
  __builtin_amdgcn_s_wait_asynccnt(0);
#else
  asm volatile("s_wait_asynccnt 0x0" ::: "memory");
#endif
#endif
}

// -------- packed-weight layout in d_ws (floats) --------
// B-operand pre-packed per WMMA k-step:
//   pair index ee = j*64 + t*32 + lane ; {W[4j+2*(lane>>4)][n], W[4j+2*(lane>>4)+1][n]}
//   n = (lane&15) + 16*t.
// WsP @ 0 (2048) | WkP @ 2048 (1024) | WqP @ 3072 | WvP @ 4096 | WcP @ 5120 (2048)
__global__ void pack_weights_kernel(const float* __restrict__ Ws,
                                    const float* __restrict__ Wk,
                                    const float* __restrict__ Wq,
                                    const float* __restrict__ Wv,
                                    const float* __restrict__ Wc,
                                    float* __restrict__ P) {
  int e = blockIdx.x * blockDim.x + threadIdx.x;
  if (e >= 3584) return;
  const float* src; int out; int ee; bool headmat;
  if (e < 1024)      { src = Ws; out = 0;    ee = e;        headmat = false; }
  else if (e < 1536) { src = Wk; out = 2048; ee = e - 1024; headmat = true;  }
  else if (e < 2048) { src = Wq; out = 3072; ee = e - 1536; headmat = true;  }
  else if (e < 2560) { src = Wv; out = 4096; ee = e - 2048; headmat = true;  }
  else               { src = Wc; out = 5120; ee = e - 2560; headmat = false; }
  int j = ee >> 6, t = (ee >> 5) & 1, l = ee & 31, hi = l >> 4;
  int n = (l & 15) + 16 * t;
  int k = 4 * j + 2 * hi;
  float v0, v1;
  if (headmat) {                      // (NH,HID,ADIM): [nh*256 + h*8 + d]
    int nh = n >> 3, d = n & 7;
    v0 = src[nh * 256 + k * 8 + d];
    v1 = src[nh * 256 + (k + 1) * 8 + d];
  } else {                            // (K,32): [k*32 + n]
    v0 = src[k * 32 + n];
    v1 = src[(k + 1) * 32 + n];
  }
  P[out + 2 * ee]     = v0;
  P[out + 2 * ee + 1] = v1;
}

__device__ __forceinline__ v8f wmma4(v2f a, v2f b, v8f c) {
  // V_WMMA_F32_16X16X4_F32 : D = A(16x4) * B(4x16) + C(16x16), full f32
  return __builtin_amdgcn_wmma_f32_16x16x4_f32(false, a, false, b, (short)0, c,
                                               false, false);
}

// C(16b x 32) += A_lds(16b x 4*ksteps, row stride astride) * packed B
__device__ __forceinline__ void gemmK(const float* __restrict__ Alds, int astride,
                                      const float* __restrict__ Bp, int ksteps,
                                      int ln, int hi, int lane, v8f& c0, v8f& c1) {
  const float* arow = Alds + ln * astride + 2 * hi;
#pragma unroll
  for (int j = 0; j < ksteps; ++j) {
    v2f a  = *(const v2f*)(arow + 4 * j);
    v2f b0 = *(const v2f*)(Bp + ((j * 2 + 0) * 32 + lane) * 2);
    v2f b1 = *(const v2f*)(Bp + ((j * 2 + 1) * 32 + lane) * 2);
    c0 = wmma4(a, b0, c0);
    c1 = wmma4(a, b1, c1);
  }
}

// stream a 16b x 64s states tile into LDS (row stride 68 floats, pad vs bank conflicts)
__device__ __forceinline__ void stage_states(const float* __restrict__ st, int a,
                                             int b0, float* __restrict__ stage,
                                             int lane) {
  const float* src = st + ((size_t)a * B_DIM + b0) * S_DIM;   // 4 KB contiguous
#if defined(USE_ASYNC_LDS)
#pragma unroll
  for (int i = 0; i < 8; ++i) {
    int g = lane + 32 * i;
    int row = g >> 4, col = (g & 15) * 4;
    __builtin_amdgcn_global_load_async_to_lds_b128(
        (GV4P)(src + g * 4), (LV4P)(stage + row * 68 + col), 0, 0);
  }
#else
#pragma unroll
  for (int i = 0; i < 8; ++i) {
    int g = lane + 32 * i;
    v4f v = __builtin_nontemporal_load((const v4f*)(src + g * 4));
    int row = g >> 4, col = (g & 15) * 4;
    *(v4f*)(stage + row * 68 + col) = v;
  }
#endif
}

__global__ __launch_bounds__(128) void critic_fused_kernel(
    const float* __restrict__ states, const float* __restrict__ bs,
    const float* __restrict__ bv, const float* __restrict__ bc,
    const float* __restrict__ P, float* __restrict__ out) {
  // per-wave region: senc(576) enc(576) stage(1088, aliased by othA) = 2240 fl
  // shared weights at 8960, 7168 fl.  Total 16128 fl = 63 KB.
  __shared__ __align__(16) float lds[16128];
  const int tid  = threadIdx.x;
  const int lane = tid & 31, wave = tid >> 5;
  const int ln = lane & 15, hi = lane >> 4;

  for (int i = tid; i < 7168 / 4; i += 128)           // weights -> LDS, once
    ((v4f*)lds)[(8960 / 4) + i] = ((const v4f*)P)[i];
  __syncthreads();

  float* wreg  = lds + wave * 2240;
  float* sencA = wreg;
  float* encA  = wreg + 576;
  float* stage = wreg + 1152;
  float* othA  = stage;                                // safe alias
  const float* WsP = lds + 8960;
  const float* WkP = WsP + 2048;
  const float* WqP = WsP + 3072;
  const float* WvP = WsP + 4096;
  const float* WcP = WsP + 5120;

  const int b0 = (blockIdx.x * 4 + wave) * 16;

  const float bs_l = bs[ln], bs_h = bs[ln + 16];
  const float bv_l = bv[ln], bv_h = bv[ln + 16];      // bv flat = n*8+d already
  const float bc_l = bc[ln], bc_h = bc[ln + 16];

  const v8f Z = {0.f, 0.f, 0.f, 0.f, 0.f, 0.f, 0.f, 0.f};

  // ---------- a = 0 : s_enc, then sel = s_enc @ Wq_all ----------
  stage_states(states, 0, b0, stage, lane);
  __builtin_prefetch(states + ((size_t)1 * B_DIM + b0) * S_DIM + lane * 32, 0, 1);
  wait_async0();
  v8f e0 = Z, e1 = Z;
  gemmK(stage, 68, WsP, 16, ln, hi, lane, e0, e1);
#pragma unroll
  for (int r = 0; r < 8; ++r) { e0[r] = LRELU(e0[r] + bs_l); e1[r] = LRELU(e1[r] + bs_h); }
#pragma unroll
  for (int r = 0; r < 8; ++r) {                        // bounce C->A layout
    int row = r + 8 * hi;
    sencA[row * 36 + ln]      = e0[r];
    sencA[row * 36 + ln + 16] = e1[r];
  }
  v8f sel0 = Z, sel1 = Z;
  gemmK(sencA, 36, WqP, 8, ln, hi, lane, sel0, sel1);
  const float scale = 0.35355339059327373f;            // 1/sqrt(ADIM=8)
#pragma unroll
  for (int r = 0; r < 8; ++r) { sel0[r] *= scale; sel1[r] *= scale; }

  // ---------- streaming attention over a = 1..15 (online softmax) ----------
  v8f acc0 = Z, acc1 = Z, s0 = Z, s1 = Z, m0, m1;
#pragma unroll
  for (int r = 0; r < 8; ++r) { m0[r] = -1e30f; m1[r] = -1e30f; }

  for (int a = 1; a < A_DIM; ++a) {
    stage_states(states, a, b0, stage, lane);
    if (a + 1 < A_DIM)                                 // pull next tile to L2
      __builtin_prefetch(states + ((size_t)(a + 1) * B_DIM + b0) * S_DIM + lane * 32, 0, 1);
    wait_async0();
    v8f c0 = Z, c1 = Z;
    gemmK(stage, 68, WsP, 16, ln, hi, lane, c0, c1);   // enc_a
#pragma unroll
    for (int r = 0; r < 8; ++r) { c0[r] = LRELU(c0[r] + bs_l); c1[r] = LRELU(c1[r] + bs_h); }
#pragma unroll
    for (int r = 0; r < 8; ++r) {
      int row = r + 8 * hi;
      encA[row * 36 + ln]      = c0[r];
      encA[row * 36 + ln + 16] = c1[r];
    }
    v8f k0 = Z, k1 = Z; gemmK(encA, 36, WkP, 8, ln, hi, lane, k0, k1);
    v8f v0 = Z, v1 = Z; gemmK(encA, 36, WvP, 8, ln, hi, lane, v0, v1);
#pragma unroll
    for (int r = 0; r < 8; ++r) { v0[r] = LRELU(v0[r] + bv_l); v1[r] = LRELU(v1[r] + bv_h); }

    // logits: reduce sel*keys over the 8 lanes of each (head,b) group,
    // result replicated per-lane -> lines up elementwise with vals
#pragma unroll
    for (int r = 0; r < 8; ++r) {
      float p = sel0[r] * k0[r];
      p += __shfl_xor(p, 1, 32); p += __shfl_xor(p, 2, 32); p += __shfl_xor(p, 4, 32);
      float mn = fmaxf(m0[r], p);
      float cr = __expf(m0[r] - mn);
      float w  = __expf(p - mn);
      s0[r]   = s0[r] * cr + w;
      acc0[r] = acc0[r] * cr + w * v0[r];
      m0[r]   = mn;

      float q = sel1[r] * k1[r];
      q += __shfl_xor(q, 1, 32); q += __shfl_xor(q, 2, 32); q += __shfl_xor(q, 4, 32);
      float mn1 = fmaxf(m1[r], q);
      float cr1 = __expf(m1[r] - mn1);
      float w1  = __expf(q - mn1);
      s1[r]   = s1[r] * cr1 + w1;
      acc1[r] = acc1[r] * cr1 + w1 * v1[r];
      m1[r]   = mn1;
    }
  }

  // other = acc/s  -> othA (aliases stage: last stage read was a=15 enc GEMM)
#pragma unroll
  for (int r = 0; r < 8; ++r) {
    int row = r + 8 * hi;
    othA[row * 36 + ln]      = acc0[r] / s0[r];
    othA[row * 36 + ln + 16] = acc1[r] / s1[r];
  }

  // ---------- critic: [s_enc | other] (16 x 64) @ Wc (64 x 32) + bc ----------
  v8f q0 = Z, q1 = Z;
  gemmK(sencA, 36, WcP,        8, ln, hi, lane, q0, q1);  // K rows  0..31
  gemmK(othA,  36, WcP + 1024, 8, ln, hi, lane, q0, q1);  // K rows 32..63
#pragma unroll
  for (int r = 0; r < 8; ++r) { q0[r] += bc_l; q1[r] += bc_h; }
#pragma unroll
  for (int r = 0; r < 8; ++r) {
    int row = b0 + r + 8 * hi;
    __builtin_nontemporal_store(q0[r], out + row * HID + ln);
    __builtin_nontemporal_store(q1[r], out + row * HID + ln + 16);
  }
}

extern "C" void kernel_launch(void* const* d_in, const int* in_sizes, int n_in,
                              void* d_out, int out_size, void* d_ws, size_t ws_size,
                              hipStream_t stream) {
  (void)in_sizes; (void)n_in; (void)out_size; (void)ws_size;
  const float* states = (const float*)d_in[0];
  const float* Ws     = (const float*)d_in[1];
  const float* bs     = (const float*)d_in[2];
  const float* Wk     = (const float*)d_in[3];
  const float* Wq     = (const float*)d_in[4];
  const float* Wv     = (const float*)d_in[5];
  const float* bv     = (const float*)d_in[6];
  const float* Wc     = (const float*)d_in[7];
  const float* bc     = (const float*)d_in[8];
  float* P = (float*)d_ws;                 // 7168 floats of packed weights

  pack_weights_kernel<<<14, 256, 0, stream>>>(Ws, Wk, Wq, Wv, Wc, P);
  // 65536 b / (4 waves * 16 b) = 1024 blocks of 128 threads (4 waves)
  critic_fused_kernel<<<1024, 128, 0, stream>>>(states, bs, bv, bc, P,
                                                (float*)d_out);
}